// BGCH_90202903150872
// MI455X (gfx1250) — compile-verified
//
#include <hip/hip_runtime.h>

typedef float v2f __attribute__((ext_vector_type(2)));
typedef float v8f __attribute__((ext_vector_type(8)));

#define N_USERS 30000
#define N_ITEMS 60000
#define N_NODES 90000
#define DIM     256
#define OUTW    768   // 3 * 256

// ---------------------------------------------------------------------------
// RP step 0: transpose v0 [256,3] (row-major) -> vk [3][256]
__global__ __launch_bounds__(256) void v0t_kernel(const float* __restrict__ v0,
                                                  float* __restrict__ vk) {
    int d = threadIdx.x;
    vk[0 * DIM + d] = v0[d * 3 + 0];
    vk[1 * DIM + d] = v0[d * 3 + 1];
    vk[2 * DIM + d] = v0[d * 3 + 2];
}

// RP step 1: t[r,c] = dot(X[r,:], vk[c,:])   (one wave32 per row)
__global__ __launch_bounds__(256) void xv_kernel(const float* __restrict__ X, int n,
                                                 const float* __restrict__ vk,
                                                 float* __restrict__ t) {
    int wave = (blockIdx.x * blockDim.x + threadIdx.x) >> 5;
    int lane = threadIdx.x & 31;
    if (wave >= n) return;
    const float* xr = X + (size_t)wave * DIM + lane * 8;
    float4 x0 = *(const float4*)(xr);
    float4 x1 = *(const float4*)(xr + 4);
    float s[3];
    #pragma unroll
    for (int c = 0; c < 3; ++c) {
        const float* vc = vk + c * DIM + lane * 8;
        float4 a = *(const float4*)(vc);
        float4 b = *(const float4*)(vc + 4);
        s[c] = x0.x * a.x + x0.y * a.y + x0.z * a.z + x0.w * a.w +
               x1.x * b.x + x1.y * b.y + x1.z * b.z + x1.w * b.w;
    }
    #pragma unroll
    for (int c = 0; c < 3; ++c)
        #pragma unroll
        for (int off = 16; off >= 1; off >>= 1)
            s[c] += __shfl_xor(s[c], off, 32);
    if (lane == 0) {
        t[(size_t)wave * 3 + 0] = s[0];
        t[(size_t)wave * 3 + 1] = s[1];
        t[(size_t)wave * 3 + 2] = s[2];
    }
}

// RP step 2: vk_out[c][d] += sum_r X[r,d] * t[r,c]   (vk_out pre-zeroed)
__global__ __launch_bounds__(256) void xtv_kernel(const float* __restrict__ X, int n,
                                                  const float* __restrict__ t,
                                                  float* __restrict__ vk_out) {
    int d = threadIdx.x;
    float a0 = 0.f, a1 = 0.f, a2 = 0.f;
    for (int r = blockIdx.x; r < n; r += gridDim.x) {
        float x  = X[(size_t)r * DIM + d];
        float t0 = t[(size_t)r * 3 + 0];
        float t1 = t[(size_t)r * 3 + 1];
        float t2 = t[(size_t)r * 3 + 2];
        a0 += x * t0; a1 += x * t1; a2 += x * t2;
    }
    atomicAdd(&vk_out[0 * DIM + d], a0);
    atomicAdd(&vk_out[1 * DIM + d], a1);
    atomicAdd(&vk_out[2 * DIM + d], a2);
}

// RP step 3: v[d] = mean_c vk[c][d]; s = sum_d v[d]^2
__global__ __launch_bounds__(256) void vfin_kernel(const float* __restrict__ vk,
                                                   float* __restrict__ v,
                                                   float* __restrict__ s) {
    int d = threadIdx.x;
    float m = (vk[d] + vk[DIM + d] + vk[2 * DIM + d]) * (1.0f / 3.0f);
    v[d] = m;
    __shared__ float sh[256];
    sh[d] = m * m;
    __syncthreads();
    for (int off = 128; off; off >>= 1) {
        if (d < off) sh[d] += sh[d + off];
        __syncthreads();
    }
    if (d == 0) *s = sh[0];
}

// RP step 4: out_row = X[r,:] - (0.5 * dot(X[r,:],v) / s) * v   (wave per row)
__global__ __launch_bounds__(256) void rp_update_kernel(const float* __restrict__ X, int n,
                                                        const float* __restrict__ v,
                                                        const float* __restrict__ s,
                                                        float* __restrict__ outbuf) {
    int wave = (blockIdx.x * blockDim.x + threadIdx.x) >> 5;
    int lane = threadIdx.x & 31;
    if (wave >= n) return;
    const float* xr = X + (size_t)wave * DIM + lane * 8;
    const float* vr = v + lane * 8;
    float4 x0 = *(const float4*)(xr);
    float4 x1 = *(const float4*)(xr + 4);
    float4 a  = *(const float4*)(vr);
    float4 b  = *(const float4*)(vr + 4);
    float p = x0.x * a.x + x0.y * a.y + x0.z * a.z + x0.w * a.w +
              x1.x * b.x + x1.y * b.y + x1.z * b.z + x1.w * b.w;
    #pragma unroll
    for (int off = 16; off >= 1; off >>= 1) p += __shfl_xor(p, off, 32);
    float coef = 0.5f * p / (*s);
    float4 o0, o1;
    o0.x = x0.x - coef * a.x; o0.y = x0.y - coef * a.y;
    o0.z = x0.z - coef * a.z; o0.w = x0.w - coef * a.w;
    o1.x = x1.x - coef * b.x; o1.y = x1.y - coef * b.y;
    o1.z = x1.z - coef * b.z; o1.w = x1.w - coef * b.w;
    float* orow = outbuf + (size_t)wave * DIM + lane * 8;
    *(float4*)(orow)     = o0;
    *(float4*)(orow + 4) = o1;
}

// ---------------------------------------------------------------------------
// Hash layer: out[:, col_off:col_off+256] = sign(A @ W^T) via V_WMMA_F32_16X16X4_F32.
// One wave computes one 16x16 C tile. A frag (16x4 f32): lane L holds
// A[row0+(L&15), k+2*(L>>4) .. +1] (v2f). B frag (4x16): B[k,n]=W[n,k], so lane L
// holds W[n0+(L&15), k+2*(L>>4) .. +1] — contiguous float2 loads from W rows.
__global__ __launch_bounds__(256) void hash_gemm_kernel(const float* __restrict__ A,
                                                        const float* __restrict__ W,
                                                        float* __restrict__ out,
                                                        int M, int col_off) {
    int wave = (blockIdx.x * blockDim.x + threadIdx.x) >> 5;
    int lane = threadIdx.x & 31;
    int row_tile = wave >> 4;   // 16 col-tiles (N=256) per row-tile
    int col_tile = wave & 15;
    if (row_tile * 16 >= M) return;
    int row0  = row_tile * 16;
    int n0    = col_tile * 16;
    int l15   = lane & 15;
    int khalf = (lane >> 4) * 2;        // 0 or 2
    const float* arow = A + (size_t)(row0 + l15) * DIM + khalf;
    const float* wrow = W + (size_t)(n0  + l15) * DIM + khalf;

    v8f acc0 = {}, acc1 = {}, acc2 = {}, acc3 = {};
    #pragma unroll 4
    for (int k = 0; k < DIM; k += 16) {
        v2f a0 = *(const v2f*)(arow + k);
        v2f b0 = *(const v2f*)(wrow + k);
        v2f a1 = *(const v2f*)(arow + k + 4);
        v2f b1 = *(const v2f*)(wrow + k + 4);
        v2f a2 = *(const v2f*)(arow + k + 8);
        v2f b2 = *(const v2f*)(wrow + k + 8);
        v2f a3 = *(const v2f*)(arow + k + 12);
        v2f b3 = *(const v2f*)(wrow + k + 12);
        acc0 = __builtin_amdgcn_wmma_f32_16x16x4_f32(false, a0, false, b0, (short)0, acc0, false, false);
        acc1 = __builtin_amdgcn_wmma_f32_16x16x4_f32(false, a1, false, b1, (short)0, acc1, false, false);
        acc2 = __builtin_amdgcn_wmma_f32_16x16x4_f32(false, a2, false, b2, (short)0, acc2, false, false);
        acc3 = __builtin_amdgcn_wmma_f32_16x16x4_f32(false, a3, false, b3, (short)0, acc3, false, false);
    }
    // D layout: VGPR i -> M = row0 + (lane>>4)*8 + i, N = n0 + (lane&15)
    int n     = n0 + l15;
    int mbase = row0 + (lane >> 4) * 8;
    #pragma unroll
    for (int i = 0; i < 8; ++i) {
        float x = acc0[i] + acc1[i] + acc2[i] + acc3[i];
        float sgn = (x > 0.f) ? 1.f : ((x < 0.f) ? -1.f : 0.f);
        out[(size_t)(mbase + i) * OUTW + col_off + n] = sgn;
    }
}

// ---------------------------------------------------------------------------
// SPMM: out[row] += val * x[col]; one wave32 per edge, 8 floats/lane,
// global_atomic_add_f32. x (92MB) + out (92MB) both resident in 192MB L2.
__global__ __launch_bounds__(256) void spmm_kernel(const int* __restrict__ erow,
                                                   const int* __restrict__ ecol,
                                                   const float* __restrict__ eval_,
                                                   const float* __restrict__ x,
                                                   float* __restrict__ out, int nedges) {
    int wid  = (blockIdx.x * blockDim.x + threadIdx.x) >> 5;
    int lane = threadIdx.x & 31;
    int nw   = (gridDim.x * blockDim.x) >> 5;
    for (int e = wid; e < nedges; e += nw) {
        int   r = erow[e];
        int   c = ecol[e];
        float w = eval_[e];
        const float* xp = x + (size_t)c * DIM + lane * 8;
        float*       op = out + (size_t)r * DIM + lane * 8;
        float4 a = *(const float4*)(xp);
        float4 b = *(const float4*)(xp + 4);
        atomicAdd(op + 0, w * a.x);
        atomicAdd(op + 1, w * a.y);
        atomicAdd(op + 2, w * a.z);
        atomicAdd(op + 3, w * a.w);
        atomicAdd(op + 4, w * b.x);
        atomicAdd(op + 5, w * b.y);
        atomicAdd(op + 6, w * b.z);
        atomicAdd(op + 7, w * b.w);
    }
}

// ---------------------------------------------------------------------------
static void run_rp(const float* X, int n, const float* v0,
                   float* bufDst,            // destination rows [n, 256]
                   float* t, float* vkA, float* vkB, float* vvec, float* svar,
                   hipStream_t stream) {
    int rowBlocks = n / 8;                   // n divisible by 8 (30000 / 60000)
    v0t_kernel<<<1, 256, 0, stream>>>(v0, vkA);
    // iter 1
    xv_kernel<<<rowBlocks, 256, 0, stream>>>(X, n, vkA, t);
    hipMemsetAsync(vkB, 0, 3 * DIM * sizeof(float), stream);
    xtv_kernel<<<1024, 256, 0, stream>>>(X, n, t, vkB);
    // iter 2
    xv_kernel<<<rowBlocks, 256, 0, stream>>>(X, n, vkB, t);
    hipMemsetAsync(vkA, 0, 3 * DIM * sizeof(float), stream);
    xtv_kernel<<<1024, 256, 0, stream>>>(X, n, t, vkA);
    // finalize + rank-1 update
    vfin_kernel<<<1, 256, 0, stream>>>(vkA, vvec, svar);
    rp_update_kernel<<<rowBlocks, 256, 0, stream>>>(X, n, vvec, svar, bufDst);
}

extern "C" void kernel_launch(void* const* d_in, const int* in_sizes, int n_in,
                              void* d_out, int out_size, void* d_ws, size_t ws_size,
                              hipStream_t stream) {
    const float* user_emb   = (const float*)d_in[0];
    const float* item_emb   = (const float*)d_in[1];
    const float* hash_W     = (const float*)d_in[2];
    const float* rp_v0_user = (const float*)d_in[3];
    const float* rp_v0_item = (const float*)d_in[4];
    const float* edge_val   = (const float*)d_in[5];
    const int*   edge_row   = (const int*)d_in[6];
    const int*   edge_col   = (const int*)d_in[7];
    float* out = (float*)d_out;
    int nedges = in_sizes[5];

    const size_t EMB = (size_t)N_NODES * DIM;   // 23,040,000 floats
    float* ws   = (float*)d_ws;
    float* bufA = ws;
    float* bufB = ws + EMB;
    float* t    = ws + 2 * EMB;                 // up to 60000*3
    float* vkA  = t + 180000;
    float* vkB  = vkA + 3 * DIM;
    float* vvec = vkB + 3 * DIM;
    float* svar = vvec + DIM;

    // 1) random projections -> bufA = concat(u, i)
    run_rp(user_emb, N_USERS, rp_v0_user, bufA, t, vkA, vkB, vvec, svar, stream);
    run_rp(item_emb, N_ITEMS, rp_v0_item, bufA + (size_t)N_USERS * DIM,
           t, vkA, vkB, vvec, svar, stream);

    const int gemmBlocks = (N_NODES / 16) * 16 / 8;   // 90000 waves / 8 per block
    const int spmmBlocks = 2048;

    // 2) layer 0 hash
    hash_gemm_kernel<<<gemmBlocks, 256, 0, stream>>>(bufA, hash_W, out, N_NODES, 0);

    // 3) spmm -> bufB, hash layer 1
    hipMemsetAsync(bufB, 0, EMB * sizeof(float), stream);
    spmm_kernel<<<spmmBlocks, 256, 0, stream>>>(edge_row, edge_col, edge_val, bufA, bufB, nedges);
    hash_gemm_kernel<<<gemmBlocks, 256, 0, stream>>>(bufB, hash_W, out, N_NODES, 256);

    // 4) spmm -> bufA, hash layer 2
    hipMemsetAsync(bufA, 0, EMB * sizeof(float), stream);
    spmm_kernel<<<spmmBlocks, 256, 0, stream>>>(edge_row, edge_col, edge_val, bufB, bufA, nedges);
    hash_gemm_kernel<<<gemmBlocks, 256, 0, stream>>>(bufA, hash_W, out, N_NODES, 512);
}